// MEFPN3D_74474732912753
// MI455X (gfx1250) — compile-verified
//
#include <hip/hip_runtime.h>
#include <hip/hip_bf16.h>
#include <math.h>

typedef __attribute__((ext_vector_type(2))) float v2f;
typedef __attribute__((ext_vector_type(8))) float v8f;

#define TPB 256

// ---------------------------------------------------------------- utilities

__global__ void k_fill(float* __restrict__ p, long n) {
  long i = (long)blockIdx.x * blockDim.x + threadIdx.x;
  if (i < n) p[i] = 0.0f;
}

// Y[idx[r]*C + c] += X[r*C + c]; idx unique within one launch -> race-free
__global__ void k_scatter_add(const float* __restrict__ X, float* __restrict__ Y,
                              const int* __restrict__ idx, long tot, int C) {
  long i = (long)blockIdx.x * blockDim.x + threadIdx.x;
  if (i < tot) {
    long r = i / C;
    long c = i - r * C;
    Y[(long)idx[r] * C + c] += X[i];
  }
}

// per-column batch stats: mu[c], rsig[c] = rsqrt(var+eps). One block per column,
// fixed partition + LDS tree reduce -> deterministic.
__global__ void k_bn_stats(const float* __restrict__ X, long N, int C,
                           float* __restrict__ mu, float* __restrict__ rs) {
  __shared__ float s1[TPB];
  __shared__ float s2[TPB];
  const int c = blockIdx.x;
  float a = 0.0f, b = 0.0f;
  for (long r = threadIdx.x; r < N; r += TPB) {
    float v = X[r * (long)C + c];
    a += v;
    b += v * v;
  }
  s1[threadIdx.x] = a;
  s2[threadIdx.x] = b;
  __syncthreads();
  for (int s = TPB / 2; s > 0; s >>= 1) {
    if ((int)threadIdx.x < s) {
      s1[threadIdx.x] += s1[threadIdx.x + s];
      s2[threadIdx.x] += s2[threadIdx.x + s];
    }
    __syncthreads();
  }
  if (threadIdx.x == 0) {
    float m = s1[0] / (float)N;
    float var = s2[0] / (float)N - m * m;
    if (var < 0.0f) var = 0.0f;
    mu[c] = m;
    rs[c] = rsqrtf(var + 1e-5f);
  }
}

// y = elu(g*(x-mu)*rsig + b)
__global__ void k_bn_elu(const float* __restrict__ X, float* __restrict__ Y,
                         const float* __restrict__ mu, const float* __restrict__ rs,
                         const float* __restrict__ gb, long tot, int C) {
  long i = (long)blockIdx.x * blockDim.x + threadIdx.x;
  if (i < tot) {
    int c = (int)(i % C);
    float v = gb[c] * (X[i] - mu[c]) * rs[c] + gb[C + c];
    Y[i] = v > 0.0f ? v : expm1f(v);
  }
}

// ---------------------------------------------------------------- WMMA GEMM
// One wave computes a 32x16 tile (two 16x16 C accumulators sharing the B
// fragment) of  Y[s(r)] (+)= X[g(r)] @ W  with V_WMMA_F32_16X16X4_F32.
// Compile-time CIN/COUT -> fully unrolled K loop, all loads base+imm-offset.
//   gidx == nullptr -> gather row r;  sidx == nullptr -> scatter row r.
// Layouts (wave32):
//   A (16x4 f32):  lane%16 = M row, K = 2*(lane/16) + v   (v = VGPR 0/1)
//   B (4x16 f32):  lane%16 = N col, K = 2*(lane/16) + v
//   C/D (16x16):   VGPR r -> M = r + 8*(lane/16), N = lane%16
template <int CIN, int COUT>
__global__ void k_gemm_tap(const float* __restrict__ X,
                           const float* __restrict__ W,   // CIN x COUT row-major
                           float* __restrict__ Y,
                           const int* __restrict__ gidx,
                           const int* __restrict__ sidx,
                           int M, int out_offset, int accumulate) {
  const int lane = threadIdx.x;                 // 0..31, one wave per y-slice
  const int tile_r = blockIdx.x * 32;
  const int tile_c = (blockIdx.y * 4 + threadIdx.y) * 16;
  const int mr = lane & 15;
  const int kh = lane >> 4;                     // K half select (0/1)

  const int r0 = tile_r + mr;
  const int r1 = tile_r + 16 + mr;
  int g0 = 0, g1 = 0;
  if (r0 < M) g0 = gidx ? gidx[r0] : r0;
  if (r1 < M) g1 = gidx ? gidx[r1] : r1;
  const float* __restrict__ xb0 = X + (long)g0 * CIN + 2 * kh;
  const float* __restrict__ xb1 = X + (long)g1 * CIN + 2 * kh;
  const float* __restrict__ wb  = W + tile_c + mr + 2 * kh * COUT;

  v8f c0 = {0.f, 0.f, 0.f, 0.f, 0.f, 0.f, 0.f, 0.f};
  v8f c1 = {0.f, 0.f, 0.f, 0.f, 0.f, 0.f, 0.f, 0.f};
#pragma unroll
  for (int k0 = 0; k0 < CIN; k0 += 4) {
    v2f a0 = *(const v2f*)(xb0 + k0);           // global_load_b64 base+imm
    v2f a1 = *(const v2f*)(xb1 + k0);
    v2f b;
    b[0] = wb[k0 * COUT];                       // imm offsets (k0*COUT*4 bytes)
    b[1] = wb[(k0 + 1) * COUT];
    c0 = __builtin_amdgcn_wmma_f32_16x16x4_f32(false, a0, false, b,
                                               (short)0, c0, false, false);
    c1 = __builtin_amdgcn_wmma_f32_16x16x4_f32(false, a1, false, b,
                                               (short)0, c1, false, false);
  }

#pragma unroll
  for (int i = 0; i < 8; ++i) {
    const int ra = tile_r + i + 8 * kh;
    if (ra < M) {
      const int ro = sidx ? sidx[ra] : ra;
      float* y = Y + (long)(ro + out_offset) * COUT + tile_c + mr;
      if (accumulate) *y += c0[i]; else *y = c0[i];
    }
    const int rb = ra + 16;
    if (rb < M) {
      const int ro = sidx ? sidx[rb] : rb;
      float* y = Y + (long)(ro + out_offset) * COUT + tile_c + mr;
      if (accumulate) *y += c1[i]; else *y = c1[i];
    }
  }
}

template <int CIN, int COUT>
static inline void gemm_t(const float* X, const float* W, float* Y,
                          const int* g, const int* s, int M, int ooff,
                          int accum, hipStream_t st) {
  dim3 grid((unsigned)((M + 31) / 32), (unsigned)(COUT / 64));
  dim3 blk(32, 4);
  k_gemm_tap<CIN, COUT><<<grid, blk, 0, st>>>(X, W, Y, g, s, M, ooff, accum);
}

static void gemm(const float* X, const float* W, float* Y,
                 const int* g, const int* s, int M, int Cin, int Cout,
                 int ooff, int accum, hipStream_t st) {
  if      (Cin == 256 && Cout == 128) gemm_t<256, 128>(X, W, Y, g, s, M, ooff, accum, st);
  else if (Cin == 128 && Cout == 128) gemm_t<128, 128>(X, W, Y, g, s, M, ooff, accum, st);
  else if (Cin == 128 && Cout ==  64) gemm_t<128,  64>(X, W, Y, g, s, M, ooff, accum, st);
  else if (Cin ==  64 && Cout ==  64) gemm_t< 64,  64>(X, W, Y, g, s, M, ooff, accum, st);
  else if (Cin ==  64 && Cout == 128) gemm_t< 64, 128>(X, W, Y, g, s, M, ooff, accum, st);
}

// ---------------------------------------------------------------- host side

extern "C" void kernel_launch(void* const* d_in, const int* in_sizes, int n_in,
                              void* d_out, int out_size, void* d_ws, size_t ws_size,
                              hipStream_t stream) {
  const float* feats0 = (const float*)d_in[0];
  const float* feats1 = (const float*)d_in[1];
  const float* feats2 = (const float*)d_in[2];
  const float* w_out0 = (const float*)d_in[3];
  const float* w_out1 = (const float*)d_in[4];
  const float* w_out2 = (const float*)d_in[5];
  const float* wt2    = (const float*)d_in[6];
  const float* wu2    = (const float*)d_in[7];
  const float* wt1    = (const float*)d_in[8];
  const float* wu1    = (const float*)d_in[9];
  const float* bn_out0 = (const float*)d_in[10];
  const float* bn_out1 = (const float*)d_in[11];
  const float* bn_out2 = (const float*)d_in[12];
  const float* bn_up2a = (const float*)d_in[13];
  const float* bn_up2b = (const float*)d_in[14];
  const float* bn_up1a = (const float*)d_in[15];
  const float* bn_up1b = (const float*)d_in[16];
  const int* m_c2 = (const int*)d_in[19];
  const int* m_g2 = (const int*)d_in[20];
  const int* m_u1 = (const int*)d_in[21];
  const int* m_g1 = (const int*)d_in[22];
  const int* m_u0 = (const int*)d_in[23];
  const int* lat1_idx = (const int*)d_in[24];
  const int* up1_idx  = (const int*)d_in[25];
  const int* lat0_idx = (const int*)d_in[26];
  const int* up0_idx  = (const int*)d_in[27];

  const int N0 = in_sizes[0] / 64;
  const int N1 = in_sizes[1] / 128;
  const int N2 = in_sizes[2] / 256;
  const int nu1 = in_sizes[17] / 128;
  const int nu0 = in_sizes[18] / 64;
  const int Mc2 = in_sizes[19] / 52;   // [2,26,M]
  const int Mg2 = in_sizes[20] / 52;
  const int Mu1 = in_sizes[21] / 52;
  const int Mg1 = in_sizes[22] / 52;
  const int Mu0 = in_sizes[23] / 52;
  const int g2n = 8 * N2;              // rows of t2
  const int g1n = 8 * nu1;             // rows of t1

  // ---- carve workspace
  char* wsp = (char*)d_ws;
  auto carve = [&](long nfloats) -> float* {
    float* p = (float*)wsp;
    wsp += ((nfloats * 4 + 255) / 256) * 256;
    return p;
  };
  long accElems = (long)(N2 + 1) * 128;
  { long e = (long)(g2n + 1) * 128; if (e > accElems) accElems = e; }
  { long e = (long)(nu1 + 1) * 128; if (e > accElems) accElems = e; }
  { long e = (long)(g1n + 1) * 64;  if (e > accElems) accElems = e; }
  { long e = (long)(nu0 + 1) * 128; if (e > accElems) accElems = e; }
  float* acc = carve(accElems);
  float* t2  = carve((long)g2n * 128);
  float* x1  = carve((long)nu1 * 128);
  float* t1  = carve((long)g1n * 64);
  float* x0  = carve((long)nu0 * 64);
  float* mu  = carve(128);
  float* rs  = carve(128);

  float* out  = (float*)d_out;
  float* out0 = out;
  float* out1 = out + (long)nu0 * 128;
  float* out2 = out + (long)nu0 * 128 + (long)nu1 * 128;

  // ---- launch helpers
  auto fill0 = [&](float* p, long n) {
    k_fill<<<dim3((unsigned)((n + TPB - 1) / TPB)), dim3(TPB), 0, stream>>>(p, n);
  };
  // full 3^3 sparse conv into acc[(n+1) x Cout]
  auto conv3 = [&](const float* X, int n, int Cin, int Cout,
                   const float* W, const int* map, int Mw) {
    fill0(acc, (long)(n + 1) * Cout);
    for (int j = 0; j < 26; ++j) {
      const int k = (j < 13) ? j : j + 1;            // skip center tap 13
      gemm(X, W + (long)k * Cin * Cout, acc,
           map + (long)j * Mw, map + (long)(26 + j) * Mw, Mw, Cin, Cout, 0, 1,
           stream);
    }
    gemm(X, W + (long)13 * Cin * Cout, acc, nullptr, nullptr, n, Cin, Cout, 0, 1,
         stream);
  };
  auto bnelu = [&](const float* src, float* dst, const float* gb, long n, int C) {
    k_bn_stats<<<dim3((unsigned)C), dim3(TPB), 0, stream>>>(src, n, C, mu, rs);
    long tot = n * C;
    k_bn_elu<<<dim3((unsigned)((tot + TPB - 1) / TPB)), dim3(TPB), 0, stream>>>(
        src, dst, mu, rs, gb, tot, C);
  };
  auto scat = [&](const float* X, float* Y, const int* idx, long rows, int C) {
    long tot = rows * C;
    k_scatter_add<<<dim3((unsigned)((tot + TPB - 1) / TPB)), dim3(TPB), 0, stream>>>(
        X, Y, idx, tot, C);
  };

  // ---------------- level 2 (top) ----------------
  // out2 = BN_ELU(sconv3(feats2, w_out2, m_c2))
  conv3(feats2, N2, 256, 128, w_out2, m_c2, Mc2);
  bnelu(acc, out2, bn_out2, N2, 128);

  // up_block_2: t2[o*N2+i] = feats2[i] @ wt2[o]; BN; conv3(wu2,m_g2); BN
  for (int o = 0; o < 8; ++o)
    gemm(feats2, wt2 + (long)o * 256 * 128, t2, nullptr, nullptr, N2, 256, 128,
         o * N2, 0, stream);
  bnelu(t2, t2, bn_up2a, g2n, 128);
  conv3(t2, g2n, 128, 128, wu2, m_g2, Mg2);
  bnelu(acc, t2, bn_up2b, g2n, 128);

  // ---------------- level 1 ----------------
  // x1 = zeros; x1[lat1]+=feats1; x1[up1]+=t2
  fill0(x1, (long)nu1 * 128);
  scat(feats1, x1, lat1_idx, N1, 128);
  scat(t2, x1, up1_idx, g2n, 128);

  conv3(x1, nu1, 128, 128, w_out1, m_u1, Mu1);
  bnelu(acc, out1, bn_out1, nu1, 128);

  // up_block_1
  for (int o = 0; o < 8; ++o)
    gemm(x1, wt1 + (long)o * 128 * 64, t1, nullptr, nullptr, nu1, 128, 64,
         o * nu1, 0, stream);
  bnelu(t1, t1, bn_up1a, g1n, 64);
  conv3(t1, g1n, 64, 64, wu1, m_g1, Mg1);
  bnelu(acc, t1, bn_up1b, g1n, 64);

  // ---------------- level 0 ----------------
  fill0(x0, (long)nu0 * 64);
  scat(feats0, x0, lat0_idx, N0, 64);
  scat(t1, x0, up0_idx, g1n, 64);

  conv3(x0, nu0, 64, 128, w_out0, m_u0, Mu0);
  bnelu(acc, out0, bn_out0, nu0, 128);

  (void)n_in; (void)out_size; (void)ws_size;
}